// RStGCN_52441550684379
// MI455X (gfx1250) — compile-verified
//
#include <hip/hip_runtime.h>
#include <math.h>

#define NN 64
#define CC 64
#define TT 512
#define VV 25
#define SS 3
#define II 16
#define TV (TT*VV)          /* 12800 */
#define COLS 256
#define NBLK (TV/COLS)      /* 50 */
#define BIG ((size_t)NN*CC*TV)

typedef float v2f __attribute__((ext_vector_type(2)));
typedef float v8f __attribute__((ext_vector_type(8)));

__device__ __forceinline__ v8f wmma_f32(v2f a, v2f b, v8f c) {
  // D = A(16x4,f32) * B(4x16,f32) + C(16x16,f32)
  return __builtin_amdgcn_wmma_f32_16x16x4_f32(false, a, false, b, (short)0, c, false, false);
}

// ---------------------------------------------------------------------------
// Tensor Data Mover staging (gfx1250). 2-D tile: tile1 rows x tile0 f32 cols,
// row stride stride0 (elements) in global, rows contiguous in LDS.
// D# layout per CDNA5 ISA.
// ---------------------------------------------------------------------------
#ifndef __has_builtin
#define __has_builtin(x) 0
#endif
#if __has_builtin(__builtin_amdgcn_tensor_load_to_lds) && \
    __has_builtin(__builtin_amdgcn_s_wait_tensorcnt)
#define HAVE_TDM 1
#else
#define HAVE_TDM 0
#endif

#if HAVE_TDM
typedef unsigned int v4u __attribute__((ext_vector_type(4)));
typedef int v4i __attribute__((ext_vector_type(4)));
typedef int v8i __attribute__((ext_vector_type(8)));

__device__ __forceinline__ void tdm_load_2d(const void* gsrc, const void* lds_dst,
                                            unsigned tile0, unsigned tile1,
                                            unsigned long long stride0)
{
  unsigned long long ga = (unsigned long long)gsrc;
  unsigned lds = (unsigned)(unsigned long long)lds_dst;  // flat LDS addr -> low 32 bits
  v4u g0 = {};
  g0.x = 1u;                                             // count = 1 (valid descriptor)
  g0.y = lds;                                            // lds_addr (bytes)
  g0.z = (unsigned)(ga & 0xffffffffull);                 // global_addr[31:0]
  g0.w = (unsigned)((ga >> 32) & 0x01ffffffull)          // global_addr[56:32]
       | 0x80000000u;                                    // type = 2 ("image")
  v8i g1 = {};
  g1[0] = 0x00020000;                                    // data_size = 4 bytes
  g1[1] = (int)((tile0 & 0xffffu) << 16);                // tensor_dim0[15:0]
  g1[2] = (int)(((tile0 >> 16) & 0xffffu)                // tensor_dim0[31:16]
              | ((tile1 & 0xffffu) << 16));              // tensor_dim1[15:0]
  g1[3] = (int)(((tile1 >> 16) & 0xffffu)                // tensor_dim1[31:16]
              | (tile0 << 16));                          // tile_dim0
  g1[4] = (int)(tile1 & 0xffffu);                        // tile_dim1 (tile_dim2 = 0)
  g1[5] = (int)(stride0 & 0xffffffffull);                // tensor_dim0_stride[31:0]
  g1[6] = (int)((stride0 >> 32) & 0xffffull);            // stride[47:32] (dim1_stride=0)
  g1[7] = 0;
  v4i gz = {};
#if __clang_major__ >= 23
  v8i gz8 = {};
  __builtin_amdgcn_tensor_load_to_lds(g0, g1, gz, gz, gz8, 0);
#else
  __builtin_amdgcn_tensor_load_to_lds(g0, g1, gz, gz, 0);
#endif
}
#endif // HAVE_TDM

// ---------------------------------------------------------------------------
// K1: adaptive adjacency  Aeff[n,s,v,w] = PA[s] + alpha*tanh( fa^T fb / (I*T) )
// fa = Wa[s] @ x[n,:,t,:] + ba ; fb likewise. One block per (n,s).
// ---------------------------------------------------------------------------
__global__ __launch_bounds__(640)
void attn_kernel(const float* __restrict__ x, const float* __restrict__ PA,
                 const float* __restrict__ alpha,
                 const float* __restrict__ Wa, const float* __restrict__ ba,
                 const float* __restrict__ Wb, const float* __restrict__ bb,
                 float* __restrict__ Aeff)
{
  __shared__ float xs[CC][VV + 1];
  __shared__ float fa[II][VV + 1];
  __shared__ float fb[II][VV + 1];
  __shared__ float Was[II][CC];
  __shared__ float Wbs[II][CC];

  const int n = blockIdx.x / SS;
  const int s = blockIdx.x % SS;
  const int tid = threadIdx.x;

  for (int j = tid; j < II * CC; j += blockDim.x) {
    Was[j / CC][j % CC] = Wa[s * II * CC + j];
    Wbs[j / CC][j % CC] = Wb[s * II * CC + j];
  }

  float g = 0.f;
  const int v = tid / VV;     // valid when tid < 625
  const int w = tid % VV;
  __syncthreads();

  for (int t = 0; t < TT; ++t) {
    __syncthreads();
    for (int j = tid; j < CC * VV; j += blockDim.x)
      xs[j / VV][j % VV] = x[(size_t)n * CC * TV + (size_t)(j / VV) * TV + t * VV + (j % VV)];
    __syncthreads();
    for (int j = tid; j < 2 * II * VV; j += blockDim.x) {
      int which = j / (II * VV);
      int r = j % (II * VV);
      int i = r / VV, vv = r % VV;
      float acc = which ? bb[s * II + i] : ba[s * II + i];
      if (which) { for (int c = 0; c < CC; ++c) acc += Wbs[i][c] * xs[c][vv]; fb[i][vv] = acc; }
      else       { for (int c = 0; c < CC; ++c) acc += Was[i][c] * xs[c][vv]; fa[i][vv] = acc; }
    }
    __syncthreads();
    if (tid < VV * VV) {
      #pragma unroll
      for (int i = 0; i < II; ++i) g += fa[i][v] * fb[i][w];
    }
  }
  if (tid < VV * VV) {
    float a1 = tanhf(g / (float)(II * TT));
    Aeff[((size_t)n * SS + s) * VV * VV + tid] = PA[s * VV * VV + tid] + alpha[0] * a1;
  }
}

// ---------------------------------------------------------------------------
// K2: y0[n,o,f] = sum_s Wd[s] @ (x[n] @ Aeff[n,s]) + sum_s bd[s]   (WMMA f32)
// block = (col-tile f0..f0+255, n); 256 thr = 8 waves, wave owns 2 n-tiles.
// x staged with TDM when the tile is fully in range.
// ---------------------------------------------------------------------------
__global__ __launch_bounds__(256)
void gcn_kernel(const float* __restrict__ x, const float* __restrict__ Aeff,
                const float* __restrict__ Wd, const float* __restrict__ bd,
                float* __restrict__ y0)
{
  __shared__ float Ae[VV][VV];
  __shared__ float WdT[CC][CC];      // WdT[c][o]
  __shared__ float xs[II][300];      // contiguous rows -> TDM tile maps 1:1
  __shared__ float xa[II][COLS];

  const int n  = blockIdx.y;
  const int f0 = blockIdx.x * COLS;
  const int tstart = f0 / VV;
  const int tid  = threadIdx.x;
  const int lane = tid & 31;
  const int wave = tid >> 5;
  const int l16  = lane & 15;
  const int lhi  = lane >> 4;        // 0|1 -> K/M half select

#if HAVE_TDM
  const bool tdm_ok = (tstart * VV + 300 <= TV);
#endif

  v8f acc[4][2] = {};

  for (int s = 0; s < SS; ++s) {
    __syncthreads();
    for (int j = tid; j < VV * VV; j += 256)
      Ae[j / VV][j % VV] = Aeff[((size_t)n * SS + s) * VV * VV + j];
    for (int j = tid; j < CC * CC; j += 256)
      WdT[j % CC][j / CC] = Wd[(size_t)s * CC * CC + j];
    __syncthreads();

    for (int ch = 0; ch < 4; ++ch) {
      __syncthreads();
#if HAVE_TDM
      if (tdm_ok) {
        if (wave == 0) {   // one TDM DMA issues the whole 16x300 tile
          tdm_load_2d(&x[(size_t)n * CC * TV + (size_t)(ch * II) * TV + tstart * VV],
                      &xs[0][0], 300u, (unsigned)II, (unsigned long long)TV);
          __builtin_amdgcn_s_wait_tensorcnt(0);
        }
      } else
#endif
      {
        for (int j = tid; j < II * 300; j += 256) {
          int c = j / 300, p = j % 300;
          int gidx = tstart * VV + p;
          xs[c][p] = (gidx < TV)
              ? x[(size_t)n * CC * TV + (size_t)(ch * II + c) * TV + gidx] : 0.f;
        }
      }
      __syncthreads();
      for (int j = tid; j < II * COLS; j += 256) {
        int c = j / COLS, col = j % COLS;
        int f = f0 + col, t = f / VV, w = f % VV;
        int toff = (t - tstart) * VV;
        float a = 0.f;
        #pragma unroll
        for (int vv = 0; vv < VV; ++vv) a += xs[c][toff + vv] * Ae[vv][w];
        xa[c][col] = a;
      }
      __syncthreads();
      #pragma unroll
      for (int nt = 0; nt < 2; ++nt) {
        int col = (wave * 2 + nt) * 16 + l16;
        #pragma unroll
        for (int k4 = 0; k4 < 4; ++k4) {
          int kb = k4 * 4 + lhi * 2;
          v2f b; b.x = xa[kb][col]; b.y = xa[kb + 1][col];
          #pragma unroll
          for (int mt = 0; mt < 4; ++mt) {
            v2f a;
            a.x = WdT[ch * II + kb][mt * 16 + l16];
            a.y = WdT[ch * II + kb + 1][mt * 16 + l16];
            acc[mt][nt] = wmma_f32(a, b, acc[mt][nt]);
          }
        }
      }
    }
  }
  #pragma unroll
  for (int mt = 0; mt < 4; ++mt)
    #pragma unroll
    for (int nt = 0; nt < 2; ++nt) {
      int col = (wave * 2 + nt) * 16 + l16;
      #pragma unroll
      for (int r = 0; r < 8; ++r) {
        int o = mt * 16 + r + lhi * 8;
        float bsum = bd[o] + bd[CC + o] + bd[2 * CC + o];
        y0[(size_t)n * CC * TV + (size_t)o * TV + f0 + col] = acc[mt][nt][r] + bsum;
      }
    }
}

// ---------------------------------------------------------------------------
// per-channel train-mode BN statistics (mean, rsqrt(var+eps))
// ---------------------------------------------------------------------------
__global__ __launch_bounds__(256)
void bnstats_kernel(const float* __restrict__ in, float* __restrict__ mean,
                    float* __restrict__ rstd)
{
  const int c = blockIdx.x;
  __shared__ float s1[256], s2[256];
  const int tid = threadIdx.x;
  float a = 0.f, b = 0.f;
  for (size_t j = tid; j < (size_t)NN * TV; j += 256) {
    size_t nn = j / TV, f = j % TV;
    float v = in[nn * CC * TV + (size_t)c * TV + f];
    a += v; b += v * v;
  }
  s1[tid] = a; s2[tid] = b; __syncthreads();
  for (int st = 128; st > 0; st >>= 1) {
    if (tid < st) { s1[tid] += s1[tid + st]; s2[tid] += s2[tid + st]; }
    __syncthreads();
  }
  if (tid == 0) {
    float inv = 1.f / (float)((size_t)NN * TV);
    float m = s1[0] * inv;
    float var = s2[0] * inv - m * m;
    mean[c] = m;
    rstd[c] = rsqrtf(var + 1e-5f);
  }
}

// BN apply + residual + relu (used for both the GCN BN and the final TCN BN)
__global__ __launch_bounds__(256)
void bnresrelu_kernel(const float* __restrict__ in, const float* __restrict__ x,
                      const float* __restrict__ mean, const float* __restrict__ rstd,
                      const float* __restrict__ g, const float* __restrict__ b,
                      float* __restrict__ out)
{
  size_t i = (size_t)blockIdx.x * 256 + threadIdx.x;
  if (i >= BIG) return;
  int c = (int)((i / TV) % CC);
  float v = (in[i] - mean[c]) * rstd[c] * g[c] + b[c] + x[i];
  out[i] = fmaxf(v, 0.f);
}

// ms[n,c,v] = mean_t y[n,c,t,v]
__global__ __launch_bounds__(256)
void meanT_kernel(const float* __restrict__ y, float* __restrict__ ms)
{
  const int n = blockIdx.x / CC, c = blockIdx.x % CC;
  __shared__ float p[8][32];
  const int tid = threadIdx.x;
  const int v = tid & 31, sl = tid >> 5;
  float a = 0.f;
  if (v < VV)
    for (int t = sl; t < TT; t += 8)
      a += y[(size_t)n * CC * TV + (size_t)c * TV + t * VV + v];
  p[sl][v] = a; __syncthreads();
  if (tid < VV) {
    float s = 0.f;
    #pragma unroll
    for (int k = 0; k < 8; ++k) s += p[k][tid];
    ms[((size_t)n * CC + c) * VV + tid] = s / (float)TT;
  }
}

// spatial attention conv1d(k=25,pad=12) -> scale 1+sigmoid
__global__ __launch_bounds__(32)
void sattn_kernel(const float* __restrict__ ms, const float* __restrict__ sw,
                  const float* __restrict__ sb, float* __restrict__ ses)
{
  const int n = blockIdx.x, v = threadIdx.x;
  if (v >= VV) return;
  float a = sb[0];
  for (int c = 0; c < CC; ++c)
    for (int j = 0; j < VV; ++j) {
      int src = v + j - 12;
      if (src >= 0 && src < VV) a += sw[c * VV + j] * ms[((size_t)n * CC + c) * VV + src];
    }
  ses[n * VV + v] = 1.f + 1.f / (1.f + expf(-a));
}

// y *= ses[n,v]; simultaneously mt[n,c,t] = mean_v y
__global__ __launch_bounds__(256)
void sscale_meanV_kernel(float* __restrict__ y, const float* __restrict__ ses,
                         float* __restrict__ mtb)
{
  int id = blockIdx.x * 256 + threadIdx.x;           // N*C*T = 2,097,152
  int n = id / (CC * TT), r = id % (CC * TT), c = r / TT, t = r % TT;
  size_t base = (size_t)n * CC * TV + (size_t)c * TV + (size_t)t * VV;
  float s = 0.f;
  #pragma unroll
  for (int v = 0; v < VV; ++v) {
    float val = y[base + v] * ses[n * VV + v];
    y[base + v] = val; s += val;
  }
  mtb[((size_t)n * CC + c) * TT + t] = s / (float)VV;
}

// temporal attention conv1d(k=9,pad=4) -> scale 1+sigmoid
__global__ __launch_bounds__(256)
void tattn_kernel(const float* __restrict__ mtb, const float* __restrict__ tw,
                  const float* __restrict__ tb, float* __restrict__ set)
{
  const int n = blockIdx.x;
  for (int t = threadIdx.x; t < TT; t += 256) {
    float a = tb[0];
    for (int c = 0; c < CC; ++c)
      #pragma unroll
      for (int j = 0; j < 9; ++j) {
        int src = t + j - 4;
        if (src >= 0 && src < TT) a += tw[c * 9 + j] * mtb[((size_t)n * CC + c) * TT + src];
      }
    set[n * TT + t] = 1.f + 1.f / (1.f + expf(-a));
  }
}

__global__ __launch_bounds__(256)
void tscale_kernel(float* __restrict__ y, const float* __restrict__ set)
{
  size_t i = (size_t)blockIdx.x * 256 + threadIdx.x;
  if (i >= BIG) return;
  int n = (int)(i / ((size_t)CC * TV));
  int t = (int)((i % TV) / VV);
  y[i] *= set[n * TT + t];
}

// mc[n,c] = mean_{t,v} y
__global__ __launch_bounds__(256)
void meanTV_kernel(const float* __restrict__ y, float* __restrict__ mc)
{
  const int n = blockIdx.x / CC, c = blockIdx.x % CC;
  __shared__ float s1[256];
  const int tid = threadIdx.x;
  float a = 0.f;
  for (int j = tid; j < TV; j += 256) a += y[(size_t)n * CC * TV + (size_t)c * TV + j];
  s1[tid] = a; __syncthreads();
  for (int st = 128; st > 0; st >>= 1) {
    if (tid < st) s1[tid] += s1[tid + st];
    __syncthreads();
  }
  if (tid == 0) mc[n * CC + c] = s1[0] / (float)TV;
}

// channel-SE MLP -> scale 1+sigmoid
__global__ __launch_bounds__(64)
void semlp_kernel(const float* __restrict__ mc, const float* __restrict__ w1,
                  const float* __restrict__ b1, const float* __restrict__ w2,
                  const float* __restrict__ b2, float* __restrict__ sec)
{
  const int n = blockIdx.x, tid = threadIdx.x;
  __shared__ float h[32];
  __shared__ float m[64];
  m[tid] = mc[n * CC + tid];
  __syncthreads();
  if (tid < 32) {
    float a = b1[tid];
    #pragma unroll
    for (int k = 0; k < CC; ++k) a += w1[tid * CC + k] * m[k];
    h[tid] = fmaxf(a, 0.f);
  }
  __syncthreads();
  float a = b2[tid];
  #pragma unroll
  for (int k = 0; k < 32; ++k) a += w2[tid * 32 + k] * h[k];
  sec[n * CC + tid] = 1.f + 1.f / (1.f + expf(-a));
}

// ---------------------------------------------------------------------------
// K13: TCN (9,1) conv as WMMA GEMM over K = C*9. Channel-SE scale is folded
// into the WEIGHT staging (w*(y*s) == (w*s)*y, s per (n,c), n fixed/block),
// so the y tile is a raw copy -> interior tiles staged with one TDM DMA.
// Flattened (t,v) -> tap shift is a flat +/-25k offset.
// ---------------------------------------------------------------------------
__global__ __launch_bounds__(256)
void tcn_kernel(const float* __restrict__ y, const float* __restrict__ tw,
                const float* __restrict__ tb, const float* __restrict__ sec,
                float* __restrict__ z)
{
  __shared__ float ys[II][456];      // halo: [f0-100, f0+356)
  __shared__ float wst[II][CC];      // wst[c_in_chunk][o] for one tap (SE-scaled)

  const int n  = blockIdx.y;
  const int f0 = blockIdx.x * COLS;
  const int tid  = threadIdx.x;
  const int lane = tid & 31;
  const int wave = tid >> 5;
  const int l16  = lane & 15;
  const int lhi  = lane >> 4;

#if HAVE_TDM
  const bool tdm_ok = (f0 >= 100) && (f0 + 356 <= TV);
#endif

  v8f acc[4][2] = {};

  for (int ch = 0; ch < 4; ++ch) {
    __syncthreads();
#if HAVE_TDM
    if (tdm_ok) {
      if (wave == 0) {   // one TDM DMA for the whole 16x456 halo tile
        tdm_load_2d(&y[(size_t)n * CC * TV + (size_t)(ch * II) * TV + (f0 - 100)],
                    &ys[0][0], 456u, (unsigned)II, (unsigned long long)TV);
        __builtin_amdgcn_s_wait_tensorcnt(0);
      }
    } else
#endif
    {
      for (int j = tid; j < II * 456; j += 256) {
        int c = j / 456, p = j % 456;
        int gidx = f0 - 100 + p;
        ys[c][p] = (gidx >= 0 && gidx < TV)
            ? y[(size_t)n * CC * TV + (size_t)(ch * II + c) * TV + gidx] : 0.f;
      }
    }
    if (ch < 3)   // warm L2 for the next channel chunk
      __builtin_prefetch(&y[(size_t)n * CC * TV + (size_t)((ch + 1) * II + (tid & 15)) * TV
                            + (f0 > 100 ? f0 - 100 : 0)], 0, 1);
    __syncthreads();

    for (int kk = 0; kk < 9; ++kk) {
      for (int j = tid; j < II * CC; j += 256) {
        int ci = j / CC, o = j % CC;
        wst[ci][o] = tw[((size_t)o * CC + (ch * II + ci)) * 9 + kk]
                   * sec[n * CC + ch * II + ci];
      }
      __syncthreads();
      const int shift = (kk - 4) * VV + 100;
      #pragma unroll
      for (int nt = 0; nt < 2; ++nt) {
        int col = (wave * 2 + nt) * 16 + l16;
        #pragma unroll
        for (int k4 = 0; k4 < 4; ++k4) {
          int kb = k4 * 4 + lhi * 2;
          v2f b; b.x = ys[kb][col + shift]; b.y = ys[kb + 1][col + shift];
          #pragma unroll
          for (int mt = 0; mt < 4; ++mt) {
            v2f a;
            a.x = wst[kb][mt * 16 + l16];
            a.y = wst[kb + 1][mt * 16 + l16];
            acc[mt][nt] = wmma_f32(a, b, acc[mt][nt]);
          }
        }
      }
      __syncthreads();
    }
  }
  #pragma unroll
  for (int mt = 0; mt < 4; ++mt)
    #pragma unroll
    for (int nt = 0; nt < 2; ++nt) {
      int col = (wave * 2 + nt) * 16 + l16;
      #pragma unroll
      for (int r = 0; r < 8; ++r) {
        int o = mt * 16 + r + lhi * 8;
        z[(size_t)n * CC * TV + (size_t)o * TV + f0 + col] = acc[mt][nt][r] + tb[o];
      }
    }
}

// ---------------------------------------------------------------------------
extern "C" void kernel_launch(void* const* d_in, const int* in_sizes, int n_in,
                              void* d_out, int out_size, void* d_ws, size_t ws_size,
                              hipStream_t stream) {
  const float* x     = (const float*)d_in[0];
  const float* PA    = (const float*)d_in[1];
  const float* alpha = (const float*)d_in[2];
  const float* Wa    = (const float*)d_in[3];
  const float* ba    = (const float*)d_in[4];
  const float* Wb    = (const float*)d_in[5];
  const float* bb    = (const float*)d_in[6];
  const float* Wd    = (const float*)d_in[7];
  const float* bd    = (const float*)d_in[8];
  const float* bn_g  = (const float*)d_in[9];
  const float* bn_b  = (const float*)d_in[10];
  const float* sa_w  = (const float*)d_in[11];
  const float* sa_b  = (const float*)d_in[12];
  const float* ta_w  = (const float*)d_in[13];
  const float* ta_b  = (const float*)d_in[14];
  const float* fc1_w = (const float*)d_in[15];
  const float* fc1_b = (const float*)d_in[16];
  const float* fc2_w = (const float*)d_in[17];
  const float* fc2_b = (const float*)d_in[18];
  const float* tcn_w = (const float*)d_in[19];
  const float* tcn_b = (const float*)d_in[20];
  const float* tbn_g = (const float*)d_in[21];
  const float* tbn_b = (const float*)d_in[22];
  float* out = (float*)d_out;

  float* ws   = (float*)d_ws;
  float* Aeff = ws;                          // 120,000
  float* big0 = ws + 131072;                 // y0, later reused as z
  float* ybuf = big0 + BIG;
  float* ms   = ybuf + BIG;                  // N*C*V
  float* mtb  = ms + (size_t)NN * CC * VV;   // N*C*T
  float* mc   = mtb + (size_t)NN * CC * TT;  // N*C
  float* ses  = mc + NN * CC;                // N*V
  float* set  = ses + NN * 32;               // N*T
  float* sec  = set + NN * TT;               // N*C
  float* mean1 = sec + NN * CC;
  float* rstd1 = mean1 + CC;
  float* mean2 = rstd1 + CC;
  float* rstd2 = mean2 + CC;

  const int EW_BLOCKS = (int)((BIG + 255) / 256);   // 204800

  attn_kernel<<<NN * SS, 640, 0, stream>>>(x, PA, alpha, Wa, ba, Wb, bb, Aeff);
  gcn_kernel<<<dim3(NBLK, NN), 256, 0, stream>>>(x, Aeff, Wd, bd, big0);
  bnstats_kernel<<<CC, 256, 0, stream>>>(big0, mean1, rstd1);
  bnresrelu_kernel<<<EW_BLOCKS, 256, 0, stream>>>(big0, x, mean1, rstd1, bn_g, bn_b, ybuf);
  meanT_kernel<<<NN * CC, 256, 0, stream>>>(ybuf, ms);
  sattn_kernel<<<NN, 32, 0, stream>>>(ms, sa_w, sa_b, ses);
  sscale_meanV_kernel<<<(NN * CC * TT) / 256, 256, 0, stream>>>(ybuf, ses, mtb);
  tattn_kernel<<<NN, 256, 0, stream>>>(mtb, ta_w, ta_b, set);
  tscale_kernel<<<EW_BLOCKS, 256, 0, stream>>>(ybuf, set);
  meanTV_kernel<<<NN * CC, 256, 0, stream>>>(ybuf, mc);
  semlp_kernel<<<NN, 64, 0, stream>>>(mc, fc1_w, fc1_b, fc2_w, fc2_b, sec);
  tcn_kernel<<<dim3(NBLK, NN), 256, 0, stream>>>(ybuf, tcn_w, tcn_b, sec, big0);
  bnstats_kernel<<<CC, 256, 0, stream>>>(big0, mean2, rstd2);
  bnresrelu_kernel<<<EW_BLOCKS, 256, 0, stream>>>(big0, x, mean2, rstd2, tbn_g, tbn_b, out);
}